// ApproachLoss_31086973288938
// MI455X (gfx1250) — compile-verified
//
#include <hip/hip_runtime.h>
#include <math.h>

// ---------------------------------------------------------------------------
// ApproachLoss for MI455X (gfx1250).
//
// Roofline: 128 MiB f32 read @ 23.3 TB/s ~= 6us, ~67 MFLOP useful -> memory
// bound. Dominant FLOPs (squared L2 norms of row differences) are mapped onto
// V_WMMA_F32_16X16X4_F32: one wave owns 16 delta rows; the A (16x4) and
// B (4x16) f32 VGPR layouts coincide when A holds diff rows and B holds the
// same diffs as columns, so we pass the SAME register pair as SRC0 and SRC1
// and accumulate C = A * A^T over K = D = 1024.  Diagonal C[m][m] is then
// ||diff_m||^2.  Diagonal lives at VGPR (m&7), lane (m<8 ? m : m+16).
// ---------------------------------------------------------------------------

typedef __attribute__((ext_vector_type(2))) float v2f;
typedef __attribute__((ext_vector_type(8))) float v8f;

#define BB 8
#define LL 4096
#define DD 1024
#define NDELT (LL - 1)      /* 4095 delta rows per batch  */
#define TILES_PER_B 256     /* ceil(4095 / 16)            */
#define WAVES_PER_BLOCK 8

__global__ __launch_bounds__(256)
void deltas_wmma_kernel(const float* __restrict__ states,
                        float* __restrict__ deltas)
{
    const int wave = blockIdx.x * WAVES_PER_BLOCK + (threadIdx.x >> 5);
    const int lane = threadIdx.x & 31;
    const int b    = wave / TILES_PER_B;
    const int tile = wave - b * TILES_PER_B;
    const int i0   = tile * 16;

    const int Lr = lane & 15;    // which of the 16 delta rows this lane feeds
    const int hf = lane >> 4;    // K half: lanes 0-15 -> K {0,1}, 16-31 -> K {2,3}

    const int di  = i0 + Lr;                             // delta row index
    const int dic = di < (NDELT - 1) ? di : (NDELT - 1); // clamp (no divergence)

    const float* __restrict__ rowc = states + ((size_t)b * LL + dic) * DD; // row i
    const float* __restrict__ rown = rowc + DD;                            // row i+1
    const int colbase = hf * 2;

    v8f c = {0.f, 0.f, 0.f, 0.f, 0.f, 0.f, 0.f, 0.f};

#pragma unroll 4
    for (int k0 = 0; k0 < DD; k0 += 4) {
        const int col = k0 + colbase;
        v2f xc = *(const v2f*)(rowc + col);   // global_load_b64
        v2f xn = *(const v2f*)(rown + col);   // global_load_b64
        v2f d;
        d.x = xn.x - xc.x;
        d.y = xn.y - xc.y;
        // D = A*B + C with A == B == d  (16x16x4 f32 WMMA, 8-arg form:
        //  neg_a, A, neg_b, B, c_mod, C, reuse_a, reuse_b)
        c = __builtin_amdgcn_wmma_f32_16x16x4_f32(
                false, d, false, d, (short)0, c, false, false);
    }

    // Pull the diagonal out of C: value for row m sits in VGPR (m&7) of C at
    // source lane (m<8 ? m : m+16).  Broadcast each and keep it on lane m.
    float nrm2 = 0.f;
#pragma unroll
    for (int m = 0; m < 16; ++m) {
        float v = __shfl(c[m & 7], (m < 8) ? m : (m + 16), 32);
        if (lane == m) nrm2 = v;
    }

    if (lane < 16 && di < NDELT) {
        deltas[b * LL + di] = sqrtf(nrm2);
    }
}

// ---------------------------------------------------------------------------
// Pass 2: tiny weighted reduction over deltas -> scalar loss.
// ---------------------------------------------------------------------------
__global__ __launch_bounds__(1024)
void loss_kernel(const float* __restrict__ deltas,
                 const float* __restrict__ reasoning_mask,
                 const int* __restrict__ result_pos,
                 float* __restrict__ out)
{
    __shared__ float s_w[1024];
    __shared__ float s_m[1024];

    const int T = LL - 2;   // 4094 delta_increase terms per batch
    float wsum = 0.f, msum = 0.f;

    for (int idx = threadIdx.x; idx < BB * T; idx += blockDim.x) {
        const int b = idx / T;
        const int t = idx - b * T;

        const float d0  = deltas[b * LL + t];
        const float d1  = deltas[b * LL + t + 1];
        float inc = d1 - d0;
        inc = inc > 0.f ? inc : 0.f;                       // relu

        const float mk = reasoning_mask[b * LL + t + 2];   // mask[:, 2:]

        int dist = result_pos[b] - t - 2;
        if (dist < 0) dist = 0;
        const float w = (dist < 5) ? (2.0f + (float)(5 - dist) * 0.5f) : 1.0f;

        wsum += inc * mk * w;
        msum += mk;
    }

    s_w[threadIdx.x] = wsum;
    s_m[threadIdx.x] = msum;
    __syncthreads();

    for (int s = blockDim.x >> 1; s > 0; s >>= 1) {
        if (threadIdx.x < (unsigned)s) {
            s_w[threadIdx.x] += s_w[threadIdx.x + s];
            s_m[threadIdx.x] += s_m[threadIdx.x + s];
        }
        __syncthreads();
    }

    if (threadIdx.x == 0) {
        out[0] = s_w[0] / (s_m[0] + 1e-9f);
    }
}

extern "C" void kernel_launch(void* const* d_in, const int* in_sizes, int n_in,
                              void* d_out, int out_size, void* d_ws, size_t ws_size,
                              hipStream_t stream)
{
    const float* states = (const float*)d_in[0];           // (8, 4096, 1024) f32
    const float* mask   = (const float*)d_in[1];           // (8, 4096) f32
    const int*   pos    = (const int*)d_in[2];             // (8,) int

    float* deltas = (float*)d_ws;                          // 8 * 4096 f32 = 128 KB

    const int total_waves = BB * TILES_PER_B;              // 2048 waves
    const int blocks      = total_waves / WAVES_PER_BLOCK; // 256 blocks x 256 thr

    deltas_wmma_kernel<<<dim3(blocks), dim3(256), 0, stream>>>(states, deltas);
    loss_kernel<<<dim3(1), dim3(1024), 0, stream>>>(deltas, mask, pos, (float*)d_out);
}